// Routing_52578989638414
// MI455X (gfx1250) — compile-verified
//
#include <hip/hip_runtime.h>
#include <math.h>
#include <stdint.h>

// ---------------------------------------------------------------------------
// MI455X (gfx1250, wave32) implementation of the CapsFPN routing reference.
//
// Input flattening (setup_inputs() dict order, tuples in (gamma, beta) order):
//   0:x  1:cW1  2:cW2  3:cW3  4,5:ln1  6,7:ln2  8,9:ln3  10,11:ln4  12,13:ln5
//   14:qW1  15,16:qln1  17:qW2  18,19:qln2  20,21:kln
// qW1/qln1/qW2/qln2_gamma are mathematically dead (LayerNorm over a size-1
// axis returns its beta), exploited exactly below.
// ---------------------------------------------------------------------------

typedef __attribute__((ext_vector_type(2))) float v2f;
typedef __attribute__((ext_vector_type(8))) float v8f;

#define LN_EPS 1e-3f
#define L2_EPS 1e-12f

__device__ __forceinline__ float eluf(float x) {
  return (x > 0.0f) ? x : expm1f(x);
}

// Async global->LDS copy, 16B per lane per issue (VGLOBAL op 98, ASYNCcnt).
// GV addressing: per-lane 64-bit global address, LDS dest = low 32 bits of the
// generic pointer.
__device__ __forceinline__ void async_copy_b128(void* lds_dst,
                                                const float* gsrc) {
  const unsigned lds_off = (unsigned)(uintptr_t)lds_dst;
  asm volatile("global_load_async_to_lds_b128 %0, %1, off" ::"v"(lds_off),
               "v"(gsrc)
               : "memory");
}

__device__ __forceinline__ void wait_async_zero() {
  asm volatile("s_wait_asynccnt 0x0" ::: "memory");
}

// ---------------------------------------------------------------------------
// condense (+ /sqrt(16) + ELU) fused with LayerNorm(8).
// GEMM: [rowsTotal x K] @ [K x 8]  (B padded to 16 cols with zeros).
// One wave per 16-row tile, V_WMMA_F32_16X16X4_F32 along K.
// A tiles staged via GLOBAL_LOAD_ASYNC_TO_LDS_B128 (no VGPR round-trip).
// ---------------------------------------------------------------------------
template <int K>
__global__ __launch_bounds__(128) void condense_ln(
    const float* __restrict__ in, long inBatchStride,
    const float* __restrict__ W,                    // [K][8] row-major
    const float* __restrict__ lng, const float* __restrict__ lnb,
    float* __restrict__ outp, long outBatchStride,  // rows of 8 floats
    int rowsPerBatch) {
  static_assert(K % 4 == 0, "K multiple of 4");
  constexpr int WAVES = 4;
  constexpr int PK = K + 4;  // pad: keeps 16B alignment, banks (4r+c)%64 distinct
  __shared__ float sA[WAVES][16][PK];
  __shared__ float sB[K][16];

  const int tid  = threadIdx.x;
  const int lane = tid & 31;
  const int w    = tid >> 5;

  const int tilesPerBatch = rowsPerBatch >> 4;
  const int tile = blockIdx.x * WAVES + w;
  const int b    = tile / tilesPerBatch;
  const int row0 = (tile % tilesPerBatch) << 4;
  const float* Abase = in + (size_t)b * inBatchStride + (size_t)row0 * K;

  // Kick off the wave-private A tile copy first: 16 x K floats, one wave-wide
  // B128 issue moves 512B.  These overlap with the sB staging below.
  constexpr int V4 = K / 4;
  for (int i = lane; i < 16 * V4; i += 32) {
    const int r = i / V4, c4 = i % V4;
    async_copy_b128(&sA[w][r][c4 * 4], Abase + (size_t)r * K + c4 * 4);
  }

  // Stage weights -> LDS, zero-pad columns 8..15 (zeros stay zero through
  // scale and ELU, so the LN sums over the 16-lane half remain exact).
  for (int i = tid; i < K * 16; i += 128) {
    const int kc = i >> 4, o = i & 15;
    sB[kc][o] = (o < 8) ? W[kc * 8 + o] : 0.0f;
  }
  __syncthreads();   // publishes sB across the block
  wait_async_zero(); // wave-private sA ready

  const int half = lane >> 4;   // K sub-pair selector per ISA A/B layouts
  const int lcol = lane & 15;   // M row (A) / N col (B,C,D)

  v8f acc = {0.f, 0.f, 0.f, 0.f, 0.f, 0.f, 0.f, 0.f};
#pragma unroll
  for (int kk = 0; kk < K; kk += 4) {
    v2f a, bb;
    a.x  = sA[w][lcol][kk + 2 * half];       // A: VGPR0 = K+0 / K+2
    a.y  = sA[w][lcol][kk + 2 * half + 1];   //    VGPR1 = K+1 / K+3
    bb.x = sB[kk + 2 * half][lcol];          // B: row-striped across lanes
    bb.y = sB[kk + 2 * half + 1][lcol];
    acc = __builtin_amdgcn_wmma_f32_16x16x4_f32(
        false, a, false, bb, (short)0, acc, false, false);
  }

  // Fused /sqrt(o*k)=/4, ELU, LayerNorm over the 8 channels of each row.
  float g = 0.0f, bias = 0.0f;
  if (lcol < 8) { g = lng[lcol]; bias = lnb[lcol]; }
  float* orow = outp + (size_t)b * outBatchStride + (size_t)row0 * 8;
#pragma unroll
  for (int e = 0; e < 8; ++e) {
    float x = eluf(acc[e] * 0.25f);          // padded cols contribute exact 0
    float s = x, s2 = x * x;
#pragma unroll
    for (int m = 1; m <= 8; m <<= 1) {       // reduce within 16-lane half
      s  += __shfl_xor(s,  m, 32);
      s2 += __shfl_xor(s2, m, 32);
    }
    const float mean = s * 0.125f;
    const float var  = s2 * 0.125f - mean * mean;
    const float inv  = rsqrtf(var + LN_EPS);
    const float o    = (x - mean) * inv * g + bias;
    if (lcol < 8) orow[(size_t)(e + 8 * half) * 8 + lcol] = o;
  }
}

// ---------------------------------------------------------------------------
// ln5 over concat rows (in place) + per-batch channel sums via LDS atomics.
// 15 blocks x 256 threads per batch (3840 rows).
// ---------------------------------------------------------------------------
__global__ __launch_bounds__(256) void ln5_colsum(
    float* __restrict__ cat, const float* __restrict__ g,
    const float* __restrict__ b, float* __restrict__ colsum) {
  __shared__ float part[8];
  const int tid = threadIdx.x;
  if (tid < 8) part[tid] = 0.0f;
  __syncthreads();

  const int batch  = blockIdx.x / 15;
  const int rowInB = (blockIdx.x % 15) * 256 + tid;
  float* row = cat + ((size_t)batch * 3840 + rowInB) * 8;

  float v[8], s = 0.f, s2 = 0.f;
#pragma unroll
  for (int d = 0; d < 8; ++d) { v[d] = row[d]; s += v[d]; s2 += v[d] * v[d]; }
  const float mean = s * 0.125f;
  const float inv  = rsqrtf(s2 * 0.125f - mean * mean + LN_EPS);
#pragma unroll
  for (int d = 0; d < 8; ++d) {
    const float o = (v[d] - mean) * inv * g[d] + b[d];
    row[d] = o;
    atomicAdd(&part[d], o);                  // ds_add_f32
  }
  __syncthreads();
  if (tid < 8) atomicAdd(&colsum[batch * 8 + tid], part[tid]);
}

// ---------------------------------------------------------------------------
// gc = l2norm(LayerNorm(colsum, kln)) per batch. 8 batches x 8 ch = 64 lanes.
// ---------------------------------------------------------------------------
__global__ void gc_kernel(const float* __restrict__ colsum,
                          const float* __restrict__ klng,
                          const float* __restrict__ klnb,
                          float* __restrict__ gcn) {
  const int t = threadIdx.x;         // 64 threads; 8-lane groups stay in-wave
  const int b = t >> 3, d = t & 7;
  const float x = colsum[b * 8 + d];
  float s = x, s2 = x * x;
#pragma unroll
  for (int m = 1; m <= 4; m <<= 1) {
    s  += __shfl_xor(s,  m, 32);
    s2 += __shfl_xor(s2, m, 32);
  }
  const float mean = s * 0.125f;
  const float inv  = rsqrtf(s2 * 0.125f - mean * mean + LN_EPS);
  const float gc   = (x - mean) * inv * klng[d] + klnb[d];
  float q = gc * gc;
#pragma unroll
  for (int m = 1; m <= 4; m <<= 1) q += __shfl_xor(q, m, 32);
  gcn[b * 8 + d] = gc * rsqrtf(fmaxf(q, L2_EPS));
}

// ---------------------------------------------------------------------------
// Final rank-1 routing: out[b,m,d] = s_k[b,m] * s_q * colsum[b,d]
//   k     = (fp[b,m]·gcn[b]) * rsqrt(max(||fp[b,m]||^2, eps))
//   s_k   = k * rsqrt(max(k^2, eps))
//   s_q   = elu(qln2_beta) normalized over its size-1 axis  (exact collapse)
// ---------------------------------------------------------------------------
__global__ __launch_bounds__(256) void routing_out(
    const float* __restrict__ cat, const float* __restrict__ gcn,
    const float* __restrict__ colsum, const float* __restrict__ qln2b,
    float* __restrict__ out) {
  const int batch  = blockIdx.x / 15;
  const int rowInB = (blockIdx.x % 15) * 256 + threadIdx.x;
  const size_t ro  = ((size_t)batch * 3840 + rowInB) * 8;
  const float* row = cat + ro;

  float dot = 0.f, nn = 0.f;
#pragma unroll
  for (int d = 0; d < 8; ++d) {
    const float v = row[d];
    dot += v * gcn[batch * 8 + d];
    nn  += v * v;
  }
  const float k  = dot * rsqrtf(fmaxf(nn, L2_EPS));
  const float sk = k * rsqrtf(fmaxf(k * k, L2_EPS));
  const float e  = eluf(qln2b[0]);
  const float sq = e * rsqrtf(fmaxf(e * e, L2_EPS));
  const float f  = sk * sq;
#pragma unroll
  for (int d = 0; d < 8; ++d) out[ro + d] = f * colsum[batch * 8 + d];
}

// ---------------------------------------------------------------------------
extern "C" void kernel_launch(void* const* d_in, const int* in_sizes, int n_in,
                              void* d_out, int out_size, void* d_ws,
                              size_t ws_size, hipStream_t stream) {
  (void)in_sizes; (void)n_in; (void)out_size; (void)ws_size;
  const float* x    = (const float*)d_in[0];
  const float* cW1  = (const float*)d_in[1];
  const float* cW2  = (const float*)d_in[2];
  const float* cW3  = (const float*)d_in[3];
  const float* ln1g = (const float*)d_in[4],  *ln1b = (const float*)d_in[5];
  const float* ln2g = (const float*)d_in[6],  *ln2b = (const float*)d_in[7];
  const float* ln3g = (const float*)d_in[8],  *ln3b = (const float*)d_in[9];
  const float* ln4g = (const float*)d_in[10], *ln4b = (const float*)d_in[11];
  const float* ln5g = (const float*)d_in[12], *ln5b = (const float*)d_in[13];
  const float* q2b  = (const float*)d_in[19];  // qln2 beta
  const float* klng = (const float*)d_in[20], *klnb = (const float*)d_in[21];

  // Workspace: cat/fp [8][3840][8] f32, then colsum[64], gcn[64].
  float* cat    = (float*)d_ws;
  float* colsum = cat + (size_t)8 * 3840 * 8;
  float* gcn    = colsum + 64;

  const long xStride   = 4096L * 64;  // per-batch floats in x
  const long catStride = 3840L * 8;   // per-batch floats in cat

  // l1..l4 into concat regions [0,2048), [2048,3072), [3072,3584), [3584,3840)
  condense_ln<128><<<256, 128, 0, stream>>>(x, xStride, cW1, ln1g, ln1b,
                                            cat, catStride, 2048);
  condense_ln<16><<<128, 128, 0, stream>>>(cat, catStride, cW2, ln2g, ln2b,
                                           cat + 2048 * 8, catStride, 1024);
  condense_ln<16><<<64, 128, 0, stream>>>(cat + 2048 * 8, catStride, cW3,
                                          ln3g, ln3b, cat + 3072 * 8,
                                          catStride, 512);
  condense_ln<16><<<32, 128, 0, stream>>>(cat + 3072 * 8, catStride, cW3,
                                          ln4g, ln4b, cat + 3584 * 8,
                                          catStride, 256);

  hipMemsetAsync(colsum, 0, 64 * sizeof(float), stream);
  ln5_colsum<<<120, 256, 0, stream>>>(cat, ln5g, ln5b, colsum);
  gc_kernel<<<1, 64, 0, stream>>>(colsum, klng, klnb, gcn);
  routing_out<<<120, 256, 0, stream>>>(cat, gcn, colsum, q2b, (float*)d_out);
}